// GCN_58703613002386
// MI455X (gfx1250) — compile-verified
//
#include <hip/hip_runtime.h>
#include <cstddef>

typedef __attribute__((ext_vector_type(2))) float v2f;
typedef __attribute__((ext_vector_type(8))) float v8f;

#define IN_C   128
#define HID_C  128
#define OUT_C  64
#define BN_EPS 1e-5f

// ---------------- degree (scatter count) ----------------
__global__ void deg_kernel(const int* __restrict__ dst, float* __restrict__ deg,
                           int E, int N) {
    int e = blockIdx.x * blockDim.x + threadIdx.x;
    if (e >= E) return;
    int d = dst[e];
    d = d < 0 ? 0 : (d >= N ? N - 1 : d);
    atomicAdd(&deg[d], 1.0f);
}

// ---------------- normalization factors ----------------
__global__ void dis_kernel(const float* __restrict__ deg, float* __restrict__ dis,
                           float* __restrict__ dis2, int N) {
    int i = blockIdx.x * blockDim.x + threadIdx.x;
    if (i >= N) return;
    float d = deg[i];
    dis[i]  = d > 0.0f ? rsqrtf(fmaxf(d, 1.0f)) : 0.0f;
    dis2[i] = rsqrtf(d + 1.0f);
}

// ---------------- sparse propagation: out[dst] += sign*ws[src]*wd[dst]*feat[src] ----
// one thread per (edge, 4-channel group); float4 gather, 4 f32 global atomics
template <int C>
__global__ void prop_kernel(const int* __restrict__ src, const int* __restrict__ dst,
                            const float* __restrict__ ws, const float* __restrict__ wd,
                            float sign, const float* __restrict__ feat,
                            float* __restrict__ out, int E, int N) {
    const int G = C / 4;
    long long tid = (long long)blockIdx.x * blockDim.x + threadIdx.x;
    if (tid >= (long long)E * G) return;
    int e = (int)(tid / G);
    int g = (int)(tid % G);
    int s = src[e]; s = s < 0 ? 0 : (s >= N ? N - 1 : s);
    int d = dst[e]; d = d < 0 ? 0 : (d >= N ? N - 1 : d);
    float w = sign * ws[s] * wd[d];
    const float4 f = *(const float4*)(feat + (long long)s * C + g * 4);
    float* o = out + (long long)d * C + g * 4;
    atomicAdd(o + 0, w * f.x);
    atomicAdd(o + 1, w * f.y);
    atomicAdd(o + 2, w * f.z);
    atomicAdd(o + 3, w * f.w);
}

// ---------------- Tx2 = 2*Tx2 - x (Chebyshev recurrence) ----------------
__global__ void cheb2_kernel(float4* __restrict__ Tx2, const float4* __restrict__ x,
                             long long n4) {
    long long i = (long long)blockIdx.x * blockDim.x + threadIdx.x;
    if (i >= n4) return;
    float4 t = Tx2[i];
    float4 xx = x[i];
    t.x = 2.0f * t.x - xx.x;
    t.y = 2.0f * t.y - xx.y;
    t.z = 2.0f * t.z - xx.z;
    t.w = 2.0f * t.w - xx.w;
    Tx2[i] = t;
}

// ---------------- Cheb GEMM (f32 WMMA) + bias + BN + ReLU ----------------
// h[N,128] = relu(BN(x@W0 + Tx1@W1 + Tx2@W2 + b)); one wave -> one 16x16 tile
__global__ __launch_bounds__(256) void cheb_gemm_kernel(
    const float* __restrict__ x, const float* __restrict__ Tx1,
    const float* __restrict__ Tx2, const float* __restrict__ W,   // [3][128][128]
    const float* __restrict__ bias, const float* __restrict__ bn_g,
    const float* __restrict__ bn_be, const float* __restrict__ bn_mu,
    const float* __restrict__ bn_var, float* __restrict__ h, int N, int rowTiles) {
    const int colTiles = HID_C / 16;                      // 8
    int wave = threadIdx.x >> 5;
    int lane = threadIdx.x & 31;
    int tile = blockIdx.x * (blockDim.x >> 5) + wave;
    if (tile >= rowTiles * colTiles) return;              // whole-wave uniform exit
    int row0 = (tile / colTiles) * 16;
    int col0 = (tile % colTiles) * 16;
    int half = lane >> 4;                                 // 0: lanes 0-15, 1: 16-31
    int l16  = lane & 15;
    int arow = row0 + l16; if (arow >= N) arow = N - 1;   // clamp, keep EXEC full
    int ncol = col0 + l16;

    v8f acc = {};
    const float* As[3] = {x, Tx1, Tx2};
    for (int m = 0; m < 3; ++m) {
        const float* arowp = As[m] + (long long)arow * IN_C;
        const float* Wm    = W + m * IN_C * HID_C;
#pragma unroll 4
        for (int k = 0; k < IN_C; k += 4) {
            int kk = k + half * 2;                        // K pair this half-wave owns
            v2f a = *(const v2f*)(arowp + kk);            // A: rows x K-pair
            v2f b;                                        // B: K-pair x cols
            b.x = Wm[kk * HID_C + ncol];
            b.y = Wm[(kk + 1) * HID_C + ncol];
            acc = __builtin_amdgcn_wmma_f32_16x16x4_f32(
                false, a, false, b, (short)0, acc, false, false);
        }
    }
    // epilogue: C/D layout -> VGPR v holds row (row0+v+8*half), col (col0+l16)
    float bcol = bias[ncol];
    float sc   = rsqrtf(bn_var[ncol] + BN_EPS) * bn_g[ncol];
    float mm   = bn_mu[ncol];
    float bb   = bn_be[ncol];
#pragma unroll
    for (int v = 0; v < 8; ++v) {
        int row = row0 + v + half * 8;
        if (row < N) {
            float val = (acc[v] + bcol - mm) * sc + bb;
            h[(long long)row * HID_C + ncol] = fmaxf(val, 0.0f);
        }
    }
}

// ---------------- GCN GEMM (f32 WMMA): hw[N,64] = h @ gcn_W ----------------
__global__ __launch_bounds__(256) void gcn_gemm_kernel(
    const float* __restrict__ h, const float* __restrict__ W,    // [128][64]
    float* __restrict__ hw, int N, int rowTiles) {
    const int colTiles = OUT_C / 16;                      // 4
    int wave = threadIdx.x >> 5;
    int lane = threadIdx.x & 31;
    int tile = blockIdx.x * (blockDim.x >> 5) + wave;
    if (tile >= rowTiles * colTiles) return;
    int row0 = (tile / colTiles) * 16;
    int col0 = (tile % colTiles) * 16;
    int half = lane >> 4;
    int l16  = lane & 15;
    int arow = row0 + l16; if (arow >= N) arow = N - 1;
    int ncol = col0 + l16;

    v8f acc = {};
    const float* arowp = h + (long long)arow * HID_C;
#pragma unroll 4
    for (int k = 0; k < HID_C; k += 4) {
        int kk = k + half * 2;
        v2f a = *(const v2f*)(arowp + kk);
        v2f b;
        b.x = W[kk * OUT_C + ncol];
        b.y = W[(kk + 1) * OUT_C + ncol];
        acc = __builtin_amdgcn_wmma_f32_16x16x4_f32(
            false, a, false, b, (short)0, acc, false, false);
    }
#pragma unroll
    for (int v = 0; v < 8; ++v) {
        int row = row0 + v + half * 8;
        if (row < N) hw[(long long)row * OUT_C + ncol] = acc[v];
    }
}

// ---------------- z = agg + (dis2^2)*hw + b; log_softmax; one wave per node ----
__global__ void finish_kernel(const float* __restrict__ agg, const float* __restrict__ hw,
                              const float* __restrict__ dis2, const float* __restrict__ bias,
                              float* __restrict__ out, int N) {
    int wave = threadIdx.x >> 5;
    int lane = threadIdx.x & 31;
    int node = blockIdx.x * (blockDim.x >> 5) + wave;
    if (node >= N) return;
    float sw = dis2[node] * dis2[node];
    long long base = (long long)node * OUT_C + lane * 2;
    const float2 a  = *(const float2*)(agg + base);
    const float2 hh = *(const float2*)(hw + base);
    float z0 = a.x + sw * hh.x + bias[lane * 2];
    float z1 = a.y + sw * hh.y + bias[lane * 2 + 1];
    float m = fmaxf(z0, z1);
    for (int off = 16; off > 0; off >>= 1) m = fmaxf(m, __shfl_xor(m, off, 32));
    float s = expf(z0 - m) + expf(z1 - m);
    for (int off = 16; off > 0; off >>= 1) s += __shfl_xor(s, off, 32);
    float lg = m + logf(s);
    float2 o; o.x = z0 - lg; o.y = z1 - lg;
    *(float2*)(out + base) = o;
}

extern "C" void kernel_launch(void* const* d_in, const int* in_sizes, int n_in,
                              void* d_out, int out_size, void* d_ws, size_t ws_size,
                              hipStream_t stream) {
    const float* x      = (const float*)d_in[0];
    const int*   ei     = (const int*)d_in[1];      // [2, E]
    const float* cheb_W = (const float*)d_in[2];    // [3,128,128]
    const float* cheb_b = (const float*)d_in[3];
    const float* bn_g   = (const float*)d_in[4];
    const float* bn_be  = (const float*)d_in[5];
    const float* bn_mu  = (const float*)d_in[6];
    const float* bn_var = (const float*)d_in[7];
    const float* gcn_W  = (const float*)d_in[8];    // [128,64]
    const float* gcn_b  = (const float*)d_in[9];

    int N = in_sizes[0] / IN_C;
    int E = in_sizes[1] / 2;
    const int* src = ei;
    const int* dst = ei + E;

    // workspace layout: [zero-block: deg|Tx1|Tx2|agg][dis|dis2|h|hw]
    float* p    = (float*)d_ws;
    float* deg  = p; p += N;
    float* Tx1  = p; p += (size_t)N * IN_C;
    float* Tx2  = p; p += (size_t)N * IN_C;
    float* agg  = p; p += (size_t)N * OUT_C;
    size_t zeroFloats = (size_t)(p - (float*)d_ws);
    float* dis  = p; p += N;
    float* dis2 = p; p += N;
    float* h    = p; p += (size_t)N * HID_C;
    float* hw   = p; p += (size_t)N * OUT_C;

    hipMemsetAsync(d_ws, 0, zeroFloats * sizeof(float), stream);

    deg_kernel<<<(E + 255) / 256, 256, 0, stream>>>(dst, deg, E, N);
    dis_kernel<<<(N + 255) / 256, 256, 0, stream>>>(deg, dis, dis2, N);

    long long wk1 = (long long)E * (IN_C / 4);
    prop_kernel<IN_C><<<(int)((wk1 + 255) / 256), 256, 0, stream>>>(
        src, dst, dis, dis, -1.0f, x, Tx1, E, N);
    prop_kernel<IN_C><<<(int)((wk1 + 255) / 256), 256, 0, stream>>>(
        src, dst, dis, dis, -1.0f, Tx1, Tx2, E, N);

    long long n4 = (long long)N * IN_C / 4;
    cheb2_kernel<<<(int)((n4 + 255) / 256), 256, 0, stream>>>(
        (float4*)Tx2, (const float4*)x, n4);

    int rowTiles = (N + 15) / 16;
    int chebTiles = rowTiles * (HID_C / 16);
    cheb_gemm_kernel<<<(chebTiles + 7) / 8, 256, 0, stream>>>(
        x, Tx1, Tx2, cheb_W, cheb_b, bn_g, bn_be, bn_mu, bn_var, h, N, rowTiles);

    int gcnTiles = rowTiles * (OUT_C / 16);
    gcn_gemm_kernel<<<(gcnTiles + 7) / 8, 256, 0, stream>>>(h, gcn_W, hw, N, rowTiles);

    long long wk3 = (long long)E * (OUT_C / 4);
    prop_kernel<OUT_C><<<(int)((wk3 + 255) / 256), 256, 0, stream>>>(
        src, dst, dis2, dis2, 1.0f, hw, agg, E, N);

    finish_kernel<<<(N + 7) / 8, 256, 0, stream>>>(agg, hw, dis2, gcn_b,
                                                   (float*)d_out, N);
}